// Projector_32220844654683
// MI455X (gfx1250) — compile-verified
//
#include <hip/hip_runtime.h>
#include <math.h>

// ---------------------------------------------------------------------------
// Cryo-EM style forward projector for MI455X (gfx1250).
//   Phase 0: build DFT basis matrices (y-iFFT and x-irDFT with all shifts,
//            Nyquist weights and 1/256 normalizations baked in).
//   Phase 1: per projection, rotate 2D coords into 3D Fourier space,
//            trilinear-gather complex weights (L2-resident tables),
//            scatter into a transposed dense half-grid PdT[kx][ky].
//   Phase 2: two dense matmuls per projection done with V_WMMA_F32_16X16X4_F32
//            (fp32 WMMA keeps FFT accuracy), mask applied in registers.
// ---------------------------------------------------------------------------

typedef __attribute__((ext_vector_type(2))) float v2f;
typedef __attribute__((ext_vector_type(8))) float v8f;

#define S_EVEN 256
#define X_HALF 129
#define KPAD   144          // 129 columns padded to 9 tiles of 16
#define CHUNK  64           // projections per workspace chunk

__device__ __forceinline__ v8f wmma4(v2f a, v2f b, v8f c) {
    // D = A(16x4) * B(4x16) + C, fp32 in/out.
    return __builtin_amdgcn_wmma_f32_16x16x4_f32(
        /*neg_a=*/false, a, /*neg_b=*/false, b,
        /*c_mod=*/(short)0, c, /*reuse_a=*/false, /*reuse_b=*/false);
}

__device__ __forceinline__ float cmaskf(int y, int x) {
    float gy = (float)y - 128.0f, gx = (float)x - 128.0f;
    float r = sqrtf(gy * gy + gx * gx);
    float t = fminf(fmaxf((128.0f - r) * 0.2f, 0.0f), 1.0f);   // (radius+e/2-r)/e
    return 0.5f - 0.5f * __cosf(3.14159265358979f * t);
}

// ---------------------------------------------------------------------------
// Phase 0: basis matrices.
//   FyR/FyI [y][j] 256x256 : (-1)^m/256 * {cos,sin}(2*pi*j*m/256), m=(y+128)&255
//   CRT/CIT [x][k] 256x144 : c_k/256 * {cos,-sin}(2*pi*k*(x^128)/256), k<=128
// ---------------------------------------------------------------------------
__global__ void init_mats(float* __restrict__ FyR, float* __restrict__ FyI,
                          float* __restrict__ CRT, float* __restrict__ CIT) {
    int i = blockIdx.x * blockDim.x + threadIdx.x;
    const float s = 1.0f / 256.0f;
    const float w = 6.28318530717958647692f / 256.0f;
    if (i < 256 * 256) {
        int y = i >> 8, j = i & 255;
        int m = (y + 128) & 255;
        float sgn = (m & 1) ? -s : s;
        float a = w * (float)((j * m) & 255);
        FyR[i] = sgn * cosf(a);
        FyI[i] = sgn * sinf(a);
    }
    if (i < 256 * KPAD) {
        int x = i / KPAD, k = i - x * KPAD;
        float vr = 0.0f, vi = 0.0f;
        if (k <= 128) {
            int x0 = x ^ 128;                                   // fftshift on x
            float ck = (k == 0 || k == 128) ? s : 2.0f * s;
            float a = w * (float)((k * x0) & 255);
            vr =  ck * cosf(a);
            vi = -ck * sinf(a);
        }
        CRT[i] = vr;
        CIT[i] = vi;
    }
}

// ---------------------------------------------------------------------------
// Phase 1: rotate + trilinear gather + fold/scatter into PdT planes.
//   PdT layout: [bl][kx(KPAD)][ky(256)] separate re/im planes (column major
//   w.r.t. the matmul so B-fragments are contiguous float2 loads).
// ---------------------------------------------------------------------------
__global__ __launch_bounds__(256)
void gather_kernel(const float* __restrict__ weight,
                   const float* __restrict__ rot,
                   const int*   __restrict__ grid,
                   const float* __restrict__ coord,
                   const int*   __restrict__ idx2d,
                   float* __restrict__ PdR, float* __restrict__ PdI,
                   int N, int b0) {
    int n = blockIdx.x * blockDim.x + threadIdx.x;
    int bl = blockIdx.y;
    if (n >= N) return;
    const float* R = rot + (size_t)(b0 + bl) * 9;   // uniform -> scalar loads
    float cx = coord[2 * n], cy = coord[2 * n + 1];
    float px = R[0] * cx + R[1] * cy;
    float py = R[3] * cx + R[4] * cy;
    float pz = R[6] * cx + R[7] * cy;
    bool flip = px < 0.0f;
    if (flip) { px = -px; py = -py; pz = -pz; }     // hermitian half
    float xf = floorf(px), yf = floorf(py), zf = floorf(pz);
    float fx = px - xf, fy = py - yf, fz = pz - zf;
    int xi = (int)xf;
    int yi = (int)yf + 131;                         // MAXR + MARGIN
    int zi = (int)zf + 131;
    float wx0 = 1.0f - fx, wy0 = 1.0f - fy, wz0 = 1.0f - fz;
    int base = (zi * 263 + yi) * 132 + xi;          // margin-padded grid dims
    float accr = 0.0f, acci = 0.0f;
#pragma unroll
    for (int dz = 0; dz < 2; ++dz) {
        float wz = dz ? fz : wz0;
#pragma unroll
        for (int dy = 0; dy < 2; ++dy) {
            float wzy = wz * (dy ? fy : wy0);
#pragma unroll
            for (int dx = 0; dx < 2; ++dx) {
                float ww = wzy * (dx ? fx : wx0);
                int wi = grid[base + dz * (263 * 132) + dy * 132 + dx];
                accr += ww * weight[2 * wi];
                acci += ww * weight[2 * wi + 1];
            }
        }
    }
    if (flip) acci = -acci;
    accr *= (1.0f / 256.0f);
    acci *= (1.0f / 256.0f);
    int idx = idx2d[n];
    int iy = idx / X_HALF;
    int ix = idx - iy * X_HALF;
    size_t dst = ((size_t)bl * KPAD + ix) * 256;
    if (iy == 0 || iy == 256) {                     // Nyquist fold into row 0
        atomicAdd(PdR + dst, 0.5f * accr);
        atomicAdd(PdI + dst, 0.5f * acci);
    } else {
        PdR[dst + iy] = accr;
        PdI[dst + iy] = acci;
    }
}

// ---------------------------------------------------------------------------
// Phase 2: per (projection, y-block) workgroup of 9 waves (288 threads).
//   Step 1: each wave computes one 16x16 complex T tile (K=256) with WMMA;
//           writes it to LDS (D-layout -> A-layout round trip).
//   Step 2: waves 0..7 each compute two 16x16 output tiles (K=144) with WMMA,
//           apply cosine mask, store.
// ---------------------------------------------------------------------------
__global__ __launch_bounds__(288)
void transform_kernel(const float* __restrict__ FyR, const float* __restrict__ FyI,
                      const float* __restrict__ CRT, const float* __restrict__ CIT,
                      const float* __restrict__ PdR, const float* __restrict__ PdI,
                      float* __restrict__ out, int b0) {
    __shared__ float TRs[16 * KPAD];
    __shared__ float TIs[16 * KPAD];
    const int bl   = blockIdx.x;
    const int yb   = blockIdx.y;           // 0..15, 16-row block of output y
    const int tid  = threadIdx.x;
    const int wave = tid >> 5;             // 0..8
    const int lane = tid & 31;
    const int lhi  = lane >> 4;            // K-pair / row-half selector
    const int llo  = lane & 15;

    const float* pdr = PdR + (size_t)bl * KPAD * 256;
    const float* pdi = PdI + (size_t)bl * KPAD * 256;
    float* outb = out + (size_t)(b0 + bl) * S_EVEN * S_EVEN;

    // ---- Step 1: T tile (rows yb*16..+15, cols wave*16..+15), K = 256 ----
    v8f tr = {}; v8f ti = {};
    {
        const int m = yb * 16 + llo;                    // A-matrix row
        const float* ar = FyR + m * 256 + 2 * lhi;
        const float* ai = FyI + m * 256 + 2 * lhi;
        const int col = wave * 16 + llo;                // B-matrix column
        const float* br = pdr + col * 256 + 2 * lhi;
        const float* bi = pdi + col * 256 + 2 * lhi;
#pragma unroll 4
        for (int kb = 0; kb < 256; kb += 4) {
            v2f aR = *(const v2f*)(ar + kb);
            v2f aI = *(const v2f*)(ai + kb);
            v2f bR = *(const v2f*)(br + kb);
            v2f bI = *(const v2f*)(bi + kb);
            v2f aIn = -aI;
            tr = wmma4(aR,  bR, tr);    // Re += FyR*PdR
            tr = wmma4(aIn, bI, tr);    // Re -= FyI*PdI
            ti = wmma4(aR,  bI, ti);    // Im += FyR*PdI
            ti = wmma4(aI,  bR, ti);    // Im += FyI*PdR
        }
    }
    // D-fragment element (row = r + 8*lhi, col = llo) -> LDS row-major T block
#pragma unroll
    for (int r = 0; r < 8; ++r) {
        int row = r + 8 * lhi;
        TRs[row * KPAD + wave * 16 + llo] = tr[r];
        TIs[row * KPAD + wave * 16 + llo] = ti[r];
    }
    __syncthreads();

    // ---- Step 2: out tiles, K = 144 (cols 129..143 of T are zero) ----
    if (wave < 8) {
        const int xt0 = wave * 2, xt1 = xt0 + 1;
        const float* aR0 = TRs + llo * KPAD + 2 * lhi;
        const float* aI0 = TIs + llo * KPAD + 2 * lhi;
        const float* b0r = CRT + (xt0 * 16 + llo) * KPAD + 2 * lhi;
        const float* b0i = CIT + (xt0 * 16 + llo) * KPAD + 2 * lhi;
        const float* b1r = CRT + (xt1 * 16 + llo) * KPAD + 2 * lhi;
        const float* b1i = CIT + (xt1 * 16 + llo) * KPAD + 2 * lhi;
        v8f acc0 = {}; v8f acc1 = {};
#pragma unroll 4
        for (int kb = 0; kb < KPAD; kb += 4) {
            v2f aR = *(const v2f*)(aR0 + kb);           // LDS (ds_load_b64)
            v2f aI = *(const v2f*)(aI0 + kb);
            v2f c0r = *(const v2f*)(b0r + kb);
            v2f c0i = *(const v2f*)(b0i + kb);
            v2f c1r = *(const v2f*)(b1r + kb);
            v2f c1i = *(const v2f*)(b1i + kb);
            acc0 = wmma4(aR, c0r, acc0);
            acc0 = wmma4(aI, c0i, acc0);
            acc1 = wmma4(aR, c1r, acc1);
            acc1 = wmma4(aI, c1i, acc1);
        }
#pragma unroll
        for (int r = 0; r < 8; ++r) {
            int y = yb * 16 + r + 8 * lhi;
            int x0 = xt0 * 16 + llo;
            int x1 = xt1 * 16 + llo;
            outb[y * S_EVEN + x0] = acc0[r] * cmaskf(y, x0);
            outb[y * S_EVEN + x1] = acc1[r] * cmaskf(y, x1);
        }
    }
}

// ---------------------------------------------------------------------------
extern "C" void kernel_launch(void* const* d_in, const int* in_sizes, int n_in,
                              void* d_out, int out_size, void* d_ws, size_t ws_size,
                              hipStream_t stream) {
    (void)n_in; (void)out_size; (void)ws_size;
    const float* weight = (const float*)d_in[0];
    const float* rot    = (const float*)d_in[1];
    const int*   grid   = (const int*)d_in[2];
    const float* coord  = (const float*)d_in[3];
    const int*   idx2d  = (const int*)d_in[4];
    float* out = (float*)d_out;

    const int B = in_sizes[1] / 9;     // 512
    const int N = in_sizes[4];         // ~25.7K in-circle 2D samples

    char* ws = (char*)d_ws;
    float* FyR = (float*)ws;  ws += (size_t)256 * 256 * sizeof(float);
    float* FyI = (float*)ws;  ws += (size_t)256 * 256 * sizeof(float);
    float* CRT = (float*)ws;  ws += (size_t)256 * KPAD * sizeof(float);
    float* CIT = (float*)ws;  ws += (size_t)256 * KPAD * sizeof(float);
    float* PdR = (float*)ws;
    float* PdI = PdR + (size_t)CHUNK * KPAD * 256;
    const size_t pd_bytes = (size_t)2 * CHUNK * KPAD * 256 * sizeof(float);

    init_mats<<<(256 * 256 + 255) / 256, 256, 0, stream>>>(FyR, FyI, CRT, CIT);

    for (int b0 = 0; b0 < B; b0 += CHUNK) {
        int nb = (B - b0 < CHUNK) ? (B - b0) : CHUNK;
        hipMemsetAsync(PdR, 0, pd_bytes, stream);
        dim3 gg((N + 255) / 256, nb);
        gather_kernel<<<gg, 256, 0, stream>>>(weight, rot, grid, coord, idx2d,
                                              PdR, PdI, N, b0);
        dim3 gt(nb, 16);
        transform_kernel<<<gt, 288, 0, stream>>>(FyR, FyI, CRT, CIT,
                                                 PdR, PdI, out, b0);
    }
}